// Decoder_2430951489916
// MI455X (gfx1250) — compile-verified
//
#include <hip/hip_runtime.h>

// Reference shapes (hardcoded per the reference file)
#define BATCH 4
#define NN    512
#define MM    512
#define ZDIM  128
#define YDIM  2
#define ROWP  132   // LDS row stride in floats (128 + 4 pad -> conflict-free column reads)

typedef float v2f __attribute__((ext_vector_type(2)));
typedef float v8f __attribute__((ext_vector_type(8)));

__global__ __launch_bounds__(512) void decoder_rbf_kernel(
    const float* __restrict__ t,      // (B, N, 1)
    const float* __restrict__ z,      // (B, N, ZD)
    const float* __restrict__ x,      // (B, M, 1)
    const float* __restrict__ sigma,  // (ZD,)
    const float* __restrict__ W,      // (YD, ZD)
    const float* __restrict__ bias,   // (YD,)
    float* __restrict__ out)          // (B, M, YD)
{
    __shared__ float tlds[NN];
    __shared__ float ylds[16 * ROWP];   // Y panel: 16 m-rows x 128 channels
    __shared__ float wlds[16 * ROWP];   // B panel: W^T zero-padded to 16 columns

    const int tid   = threadIdx.x;
    const int lane  = tid & 31;
    const int wave  = tid >> 5;           // 0..15 -> row of the m-tile
    const int b     = blockIdx.x >> 5;    // 32 m-tiles per batch
    const int mtile = blockIdx.x & 31;
    const int m     = mtile * 16 + wave;

    // ---- One-shot staging (all disjoint, single barrier) ----
    // t[b, :] -> LDS (512 threads, 512 values)
    tlds[tid] = t[b * NN + tid];
    // Zero-fill the padded W panel...
    for (int i = tid; i < 16 * ROWP; i += 512) wlds[i] = 0.0f;
    // ...then drop in the two live rows: wlds[y][c] = W[y, c]
    if (tid < YDIM * ZDIM) {
        const int yrow = tid >> 7;        // /ZDIM
        const int c    = tid & (ZDIM - 1);
        wlds[yrow * ROWP + c] = W[yrow * ZDIM + c];
    }
    __syncthreads();

    // ---- Main transcendental-bound reduction ----
    // Each lane owns 4 consecutive channels: c0..c0+3 (wave covers all 128).
    const int   c0 = lane * 4;
    const float q0 = -0.5f * __expf(-2.0f * sigma[c0 + 0]);
    const float q1 = -0.5f * __expf(-2.0f * sigma[c0 + 1]);
    const float q2 = -0.5f * __expf(-2.0f * sigma[c0 + 2]);
    const float q3 = -0.5f * __expf(-2.0f * sigma[c0 + 3]);
    const float xm = x[b * MM + m];

    // z[b, n, c0..c0+3]: contiguous float4 per lane, 512B/wave per n (coalesced)
    const float4* zp = (const float4*)(z + (size_t)b * NN * ZDIM + c0);

    float a0 = 0.f, a1 = 0.f, a2 = 0.f, a3 = 0.f;

    // Lane-uniform fast path: all 4 channel scales equal -> 1 exp serves 4 channels.
    if (q0 == q1 && q1 == q2 && q2 == q3) {
        #pragma unroll 4
        for (int n = 0; n < NN; ++n) {
            const float d = xm - tlds[n];
            const float e = __expf(d * d * q0);
            const float4 zv = zp[n * (ZDIM / 4)];
            a0 = __builtin_fmaf(zv.x, e, a0);
            a1 = __builtin_fmaf(zv.y, e, a1);
            a2 = __builtin_fmaf(zv.z, e, a2);
            a3 = __builtin_fmaf(zv.w, e, a3);
        }
    } else {
        #pragma unroll 4
        for (int n = 0; n < NN; ++n) {
            const float d = xm - tlds[n];
            const float u = d * d;
            const float4 zv = zp[n * (ZDIM / 4)];
            a0 = __builtin_fmaf(zv.x, __expf(u * q0), a0);
            a1 = __builtin_fmaf(zv.y, __expf(u * q1), a1);
            a2 = __builtin_fmaf(zv.z, __expf(u * q2), a2);
            a3 = __builtin_fmaf(zv.w, __expf(u * q3), a3);
        }
    }

    // Park Y[16 rows of m, 128 channels] in LDS for the WMMA projection.
    {
        float* yr = &ylds[wave * ROWP + c0];
        yr[0] = a0; yr[1] = a1; yr[2] = a2; yr[3] = a3;
    }
    __syncthreads();

    // ---- Wave 0: out[16m, 16y(pad)] = Y(16x128) @ W^T via 32x V_WMMA_F32_16X16X4_F32 ----
    if (wave == 0) {
        // ISA 16x4 f32 A layout: VGPR0 = {K=0 | K=2}, VGPR1 = {K=1 | K=3},
        // lanes 0-15 -> first K pair, lanes 16-31 -> second; M = lane & 15.
        const int mrow = lane & 15;
        const int koff = (lane >> 4) * 2;   // 0 or 2
        const int ncol = lane & 15;         // B/N column index

        const float* ya = &ylds[mrow * ROWP + koff];
        const float* wb = &wlds[ncol * ROWP + koff];

        v8f acc = {};                       // C starts at 0
        #pragma unroll
        for (int kk = 0; kk < ZDIM / 4; ++kk) {
            const int cb = kk * 4;
            v2f A, Bm;
            A.x = ya[cb + 0];               // -> ds_load_b64
            A.y = ya[cb + 1];
            Bm.x = wb[cb + 0];              // -> ds_load_b64 (zero-padded panel)
            Bm.y = wb[cb + 1];
            acc = __builtin_amdgcn_wmma_f32_16x16x4_f32(
                false, A, false, Bm, (short)0, acc, false, false);
        }
        // D layout: VGPR r -> M=r (lanes 0-15), M=r+8 (lanes 16-31); N = lane&15.
        const int nout  = lane & 15;
        const int mhalf = (lane >> 4) * 8;
        if (nout < YDIM) {
            const float bv = bias[nout];
            #pragma unroll
            for (int r = 0; r < 8; ++r) {
                const int mo = mtile * 16 + mhalf + r;
                out[((size_t)b * MM + mo) * YDIM + nout] = acc[r] + bv;
            }
        }
    }
}

extern "C" void kernel_launch(void* const* d_in, const int* in_sizes, int n_in,
                              void* d_out, int out_size, void* d_ws, size_t ws_size,
                              hipStream_t stream) {
    (void)in_sizes; (void)n_in; (void)d_ws; (void)ws_size; (void)out_size;
    const float* t     = (const float*)d_in[0];
    const float* z     = (const float*)d_in[1];
    const float* x     = (const float*)d_in[2];
    const float* sigma = (const float*)d_in[3];
    const float* W     = (const float*)d_in[4];
    const float* bias  = (const float*)d_in[5];
    float* out = (float*)d_out;

    dim3 grid(BATCH * (MM / 16));   // 128 blocks: (batch, m-tile)
    dim3 block(512);                // 16 waves of 32
    decoder_rbf_kernel<<<grid, block, 0, stream>>>(t, z, x, sigma, W, bias, out);
}